// Lifter_51565377356545
// MI455X (gfx1250) — compile-verified
//
#include <hip/hip_runtime.h>

typedef __attribute__((ext_vector_type(2))) float v2f;
typedef __attribute__((ext_vector_type(8))) float v8f;
typedef int v4i_vs __attribute__((vector_size(4 * sizeof(int))));  // matches builtin param type

#define C_DIM 128
#define D_DIM 128
#define ROWS_PER_BLOCK 128
#define STRT 132   // vox tile row stride (floats): 132 mod 64 = 4 -> banks spread across rows
#define STR2 144   // W^T pair-row stride (float2): 2*144 mod 64 = 32 -> half-waves hit disjoint banks

#if defined(__AMDGCN__) && __has_builtin(__builtin_amdgcn_global_load_async_to_lds_b128) && \
    __has_builtin(__builtin_amdgcn_s_wait_asynccnt)
#define USE_ASYNC_LDS 1
#else
#define USE_ASYNC_LDS 0
#endif

// ---------------------------------------------------------------------------
// Zero helpers
// ---------------------------------------------------------------------------
__global__ void zero4_kernel(float4* __restrict__ p, long n4) {
    long i = (long)blockIdx.x * blockDim.x + threadIdx.x;
    if (i < n4) p[i] = make_float4(0.f, 0.f, 0.f, 0.f);
}

__global__ void zero1_kernel(float* __restrict__ p, long n) {
    long i = (long)blockIdx.x * blockDim.x + threadIdx.x;
    if (i < n) p[i] = 0.f;
}

// ---------------------------------------------------------------------------
// Scatter: one thread per (pixel, channel). Feature loads fully coalesced.
// The 205 MB sums accumulator is L2-resident (192 MB L2), so fp32 global
// atomics (global_atomic_add_f32) stay on-chip.
// ---------------------------------------------------------------------------
__global__ __launch_bounds__(256) void scatter_kernel(
    const float* __restrict__ feats,   // [B*P, C]
    const int*   __restrict__ vids,    // [B*P]
    const int*   __restrict__ hit,     // [B*P]
    const int*   __restrict__ eff,     // [B*P]
    float*       __restrict__ sums,    // [B*V, C]  (aliases d_out)
    float*       __restrict__ counts,  // [B*V]
    int P, int V, long n_elems)
{
    long gid = (long)blockIdx.x * blockDim.x + threadIdx.x;
    if (gid >= n_elems) return;
    int  col = (int)(gid & (C_DIM - 1));
    long pix = gid >> 7;
    if (hit[pix] == 0 || eff[pix] == 0) return;        // masked -> dump segment
    int  b   = (int)(pix / P);
    long seg = (long)vids[pix] + (long)b * V;
    unsafeAtomicAdd(&sums[seg * C_DIM + col], feats[gid]);
    if (col == 0) unsafeAtomicAdd(&counts[seg], 1.0f);
}

// ---------------------------------------------------------------------------
// Fused mean-divide + (vox @ W^T + b), in place over d_out.
// 8 waves/block; wave w owns 16-row band. fp32 WMMA 16x16x4 (exact fp32,
// phase is HBM-bound anyway, ~18us floor at 23.3 TB/s).
//
// A frag (16x4 f32, ISA 7.12.2): lanes 0-15 K=0,1; lanes 16-31 K=2,3;
// M = lane&15. B frag mirrored with N = lane&15. C/D: vgpr i holds row
// (i + 8*(lane>=16)), col = lane&15.
//
// W^T is staged pair-major: lds_wt2[k/2][n] = {W[n][k], W[n][k+1]} so each
// B fragment is one even-aligned ds_load_b64 (no repack moves).
// ---------------------------------------------------------------------------
__global__ __launch_bounds__(256) void mix_fc_kernel(
    float*       __restrict__ vox,     // in: sums, out: result  [rows, 128]
    const float* __restrict__ counts,  // [rows]
    const float* __restrict__ W,       // [D, C] row-major
    const float* __restrict__ bias,    // [D]
    int rows)
{
    __shared__ float2 lds_wt2[(C_DIM / 2) * STR2];    // 73.7 KB  W^T pair-major
    __shared__ float  lds_t[ROWS_PER_BLOCK * STRT];   // 67.6 KB  vox tile [r][c]
    __shared__ float  lds_inv[ROWS_PER_BLOCK];
    __shared__ float  lds_bias[D_DIM];

    const int tid  = threadIdx.x;
    const int row0 = blockIdx.x * ROWS_PER_BLOCK;

    // Stage W^T pair-major. W is 64 KB, block-reused 128x -> L2-cached reads.
    for (int i = tid; i < (C_DIM / 2) * D_DIM; i += 256) {
        int kp = i >> 7;          // K-pair index 0..63
        int n  = i & 127;         // output-channel index
        lds_wt2[kp * STR2 + n] = make_float2(W[n * C_DIM + 2 * kp],
                                             W[n * C_DIM + 2 * kp + 1]);
    }

    // Stage vox tile: async global->LDS (ASYNCcnt path) when available.
#if USE_ASYNC_LDS
    for (int i = tid; i < ROWS_PER_BLOCK * (C_DIM / 4); i += 256) {
        int r  = i >> 5;                 // 32 16-byte chunks per row
        int c4 = (i & 31) * 4;
        int gr = row0 + r;
        float* ldst = &lds_t[r * STRT + c4];
        if (gr < rows) {
            __builtin_amdgcn_global_load_async_to_lds_b128(
                (v4i_vs*)(vox + (long)gr * C_DIM + c4), (v4i_vs*)ldst, 0, 0);
        } else {
            ldst[0] = 0.f; ldst[1] = 0.f; ldst[2] = 0.f; ldst[3] = 0.f;
        }
    }
    __builtin_amdgcn_s_wait_asynccnt(0);
#else
    for (int i = tid; i < ROWS_PER_BLOCK * C_DIM; i += 256) {
        int r = i >> 7, c = i & 127;
        int gr = row0 + r;
        lds_t[r * STRT + c] = (gr < rows) ? vox[(long)gr * C_DIM + c] : 0.f;
    }
#endif

    if (tid < ROWS_PER_BLOCK) {
        int gr = row0 + tid;
        float cnt = (gr < rows) ? counts[gr] : 1.f;
        lds_inv[tid] = 1.0f / fmaxf(cnt, 1.0f);
    }
    if (tid < D_DIM) lds_bias[tid] = bias[tid];
    __syncthreads();

    const int wave = tid >> 5;
    const int lane = tid & 31;
    const int half = lane >> 4;        // 0: lanes 0-15, 1: lanes 16-31
    const int l16  = lane & 15;
    const int m0   = wave * 16;        // wave's row band within tile

    v8f acc[8] = {};                   // 8 N-tiles of 16x16 f32

    const float  inv  = lds_inv[m0 + l16];
    const float* tRow = &lds_t[(m0 + l16) * STRT];

    for (int k = 0; k < C_DIM; k += 4) {
        const int ka = k + half * 2;             // even -> 8B-aligned LDS reads
        const float2 av = *(const float2*)&tRow[ka];
        v2f a;
        a.x = av.x * inv;
        a.y = av.y * inv;
        const float2* brow = &lds_wt2[(ka >> 1) * STR2];
        #pragma unroll
        for (int n = 0; n < 8; ++n) {
            const float2 bv = brow[n * 16 + l16];  // single ds_load_b64
            v2f bf;
            bf.x = bv.x;
            bf.y = bv.y;
            acc[n] = __builtin_amdgcn_wmma_f32_16x16x4_f32(
                false, a, false, bf, (short)0, acc[n], false, false);
        }
    }

    // Bias add + in-place store (each wave's band is only re-read from LDS).
    #pragma unroll
    for (int n = 0; n < 8; ++n) {
        const int col = n * 16 + l16;
        const float bv = lds_bias[col];
        #pragma unroll
        for (int i = 0; i < 8; ++i) {
            int r = row0 + m0 + i + half * 8;
            if (r < rows) vox[(long)r * D_DIM + col] = acc[n][i] + bv;
        }
    }
}

// ---------------------------------------------------------------------------
extern "C" void kernel_launch(void* const* d_in, const int* in_sizes, int n_in,
                              void* d_out, int out_size, void* d_ws, size_t ws_size,
                              hipStream_t stream) {
    const float* feats = (const float*)d_in[0];
    const float* W     = (const float*)d_in[1];
    const float* bias  = (const float*)d_in[2];
    const int*   vids  = (const int*)d_in[3];
    const int*   hit   = (const int*)d_in[4];
    const int*   eff   = (const int*)d_in[5];
    // d_in[6] = total_voxels on device; derive V from sizes instead.

    const int Bb = 2;
    long BP = in_sizes[3];                 // B*P
    int  P  = (int)(BP / Bb);
    long BV = (long)out_size / D_DIM;      // B*V rows
    int  V  = (int)(BV / Bb);
    long NE = BP * (long)C_DIM;            // total feature elements

    float* sums   = (float*)d_out;         // accumulate in place
    float* counts = (float*)d_ws;          // B*V floats (1.6 MB)

    // 1) zero sums (d_out) and counts
    {
        long n4 = (long)out_size / 4;
        zero4_kernel<<<(int)((n4 + 255) / 256), 256, 0, stream>>>((float4*)d_out, n4);
        zero1_kernel<<<(int)((BV + 255) / 256), 256, 0, stream>>>(counts, BV);
    }
    // 2) masked scatter-add of features + counts
    scatter_kernel<<<(int)((NE + 255) / 256), 256, 0, stream>>>(
        feats, vids, hit, eff, sums, counts, P, V, NE);
    // 3) fused mean-divide + GEMM (+bias), in place
    mix_fc_kernel<<<(int)((BV + ROWS_PER_BLOCK - 1) / ROWS_PER_BLOCK), 256, 0, stream>>>(
        sums, counts, W, bias, (int)BV);
}